// RGCNConv_27779848471358
// MI455X (gfx1250) — compile-verified
//
#include <hip/hip_runtime.h>

#define CDIM    128
#define RREL    8
#define OUTD    128
#define KDIM    (CDIM * RREL)          // 1024
#define TPB     512                    // 16 wave32s
#define TILE_M  32                     // targets per workgroup
#define ROW_STRIDE 1156                // 1024 agg + 128 root + 4 pad (stride % 64 == 4 -> LDS conflict-free)
#define AGG_FLOATS (TILE_M * ROW_STRIDE)
#define CNT_OFF    AGG_FLOATS
#define LDS_FLOATS (AGG_FLOATS + TILE_M * RREL)

typedef __attribute__((ext_vector_type(2))) float v2f;
typedef __attribute__((ext_vector_type(8))) float v8f;

__global__ __launch_bounds__(TPB) void rgcn_fused_kernel(
    const float* __restrict__ x,          // [T + N_NEIGH][C]
    const int*   __restrict__ row_ptr,    // [T+1]
    const int*   __restrict__ col_ind,    // [E] -> neighbor index
    const int*   __restrict__ etype,      // [E]
    const float* __restrict__ lin_w,      // [OUT][KDIM]
    const float* __restrict__ root_w,     // [OUT][C]
    const float* __restrict__ root_b,     // [OUT]
    float*       __restrict__ y,          // [T][OUT]
    int num_targets)
{
    extern __shared__ float lds[];
    const int tid  = threadIdx.x;
    const int lane = tid & 31;           // wave32
    const int wave = tid >> 5;           // 0..15
    const int t0   = blockIdx.x * TILE_M;

    // ---- zero LDS (agg + root region + counts) ----
    for (int i = tid; i < LDS_FLOATS; i += TPB) lds[i] = 0.0f;
    __syncthreads();

    // ---- Phase 1: gather neighbors, accumulate per (target, relation) ----
    // wave w exclusively owns local targets 2w and 2w+1.
    #pragma unroll
    for (int s = 0; s < 2; ++s) {
        const int lt = wave * 2 + s;
        const int t  = t0 + lt;
        if (t < num_targets) {
            const int e0 = row_ptr[t];
            const int e1 = row_ptr[t + 1];
            for (int e = e0; e < e1; ++e) {
                const int col = col_ind[e];
                const int rel = etype[e];
                const float4 v = *(const float4*)(x + (size_t)(num_targets + col) * CDIM + lane * 4);
                float* dst = &lds[lt * ROW_STRIDE + rel * CDIM + lane * 4];
                atomicAdd(dst + 0, v.x);
                atomicAdd(dst + 1, v.y);
                atomicAdd(dst + 2, v.z);
                atomicAdd(dst + 3, v.w);
                if (lane == 0) atomicAdd(&lds[CNT_OFF + lt * RREL + rel], 1.0f);
            }
            // append root (target) features as extra K columns [1024, 1152)
            const float4 xr = *(const float4*)(x + (size_t)t * CDIM + lane * 4);
            *(float4*)&lds[lt * ROW_STRIDE + KDIM + lane * 4] = xr;
        }
        // if t >= num_targets the row stays zero from the init pass
    }
    __syncthreads();

    // ---- Phase 2: counts -> reciprocal of max(cnt, 1) ----
    if (tid < TILE_M * RREL) {
        float c = lds[CNT_OFF + tid];
        lds[CNT_OFF + tid] = 1.0f / fmaxf(c, 1.0f);
    }
    __syncthreads();

    // ---- Phase 3: scale sums -> means ----
    for (int i = tid; i < TILE_M * KDIM; i += TPB) {
        const int lt = i >> 10;          // / 1024
        const int kc = i & 1023;
        const float sc = lds[CNT_OFF + lt * RREL + (kc >> 7)];
        lds[lt * ROW_STRIDE + kc] *= sc;
    }
    __syncthreads();

    // ---- Phase 4: fused GEMM  y = agg @ lin_w^T + x_t @ root_w^T + b ----
    // 16 waves tile the 32x128 output as (2 m-tiles) x (8 n-tiles) of 16x16.
    const int m_tile = wave >> 3;        // 0..1
    const int n_tile = wave & 7;         // 0..7
    const int mn     = lane & 15;        // A: row m ; B: col n
    const int g      = lane >> 4;        // K sub-group

    v8f acc = {};
    const float* a_base = &lds[(size_t)(m_tile * 16 + mn) * ROW_STRIDE + 2 * g];
    const float* b_base = lin_w + (size_t)(n_tile * 16 + mn) * KDIM + 2 * g;

    // main GEMM over aggregated features: K = 1024
    #pragma unroll 4
    for (int k0 = 0; k0 < KDIM; k0 += 4) {
        v2f a = *(const v2f*)(a_base + k0);          // ds_load_b64, conflict-free
        v2f b = *(const v2f*)(b_base + k0);          // global_load_b64 (L2 resident)
        acc = __builtin_amdgcn_wmma_f32_16x16x4_f32(
            false, a, false, b, (short)0, acc, false, false);
    }

    // root GEMM: K = 128, A comes from the appended columns
    const float* a2 = a_base + KDIM;
    const float* b2 = root_w + (size_t)(n_tile * 16 + mn) * CDIM + 2 * g;
    #pragma unroll 4
    for (int k0 = 0; k0 < CDIM; k0 += 4) {
        v2f a = *(const v2f*)(a2 + k0);
        v2f b = *(const v2f*)(b2 + k0);
        acc = __builtin_amdgcn_wmma_f32_16x16x4_f32(
            false, a, false, b, (short)0, acc, false, false);
    }

    // ---- bias + store: D layout VGPR r, lane l -> M = r + 8*(l>=16), N = l&15 ----
    const float bias = root_b[n_tile * 16 + mn];
    #pragma unroll
    for (int r = 0; r < 8; ++r) {
        const int m  = m_tile * 16 + r + 8 * g;
        const int gt = t0 + m;
        if (gt < num_targets)
            y[(size_t)gt * OUTD + n_tile * 16 + mn] = acc[r] + bias;
    }
}

extern "C" void kernel_launch(void* const* d_in, const int* in_sizes, int n_in,
                              void* d_out, int out_size, void* d_ws, size_t ws_size,
                              hipStream_t stream) {
    const float* x        = (const float*)d_in[0];
    const int*   row_ptr  = (const int*)  d_in[1];
    const int*   col_ind  = (const int*)  d_in[2];
    const int*   etype    = (const int*)  d_in[3];
    const float* lin_w    = (const float*)d_in[4];
    const float* root_w   = (const float*)d_in[5];
    const float* root_b   = (const float*)d_in[6];
    float*       y        = (float*)d_out;

    const int num_targets = in_sizes[1] - 1;   // row_ptr has T+1 entries
    const int grid = (num_targets + TILE_M - 1) / TILE_M;
    const size_t shmem = (size_t)LDS_FLOATS * sizeof(float);  // ~145.5 KB (dyn LDS; WGP has 320 KB)

    (void)hipFuncSetAttribute((const void*)rgcn_fused_kernel,
                              hipFuncAttributeMaxDynamicSharedMemorySize, (int)shmem);

    rgcn_fused_kernel<<<grid, TPB, shmem, stream>>>(
        x, row_ptr, col_ind, etype, lin_w, root_w, root_b, y, num_targets);
}